// GraphAttentionLayer_36455682408927
// MI455X (gfx1250) — compile-verified
//
#include <hip/hip_runtime.h>

typedef float v2f __attribute__((ext_vector_type(2)));
typedef float v8f __attribute__((ext_vector_type(8)));

#define N_NODES 8192
#define F_IN    256
#define F_OUT   64
#define ALPHA   0.2f
#define KC      128            // K columns staged in LDS per chunk
#define NROWS   (KC / 2)       // paired-K rows in LDS
#define RSTRIDE 144            // floats per paired row; 144%64=16, 288%64=32 -> conflict-free
#define WAVES_PER_WG 8
#define ROWS_PER_WAVE 16

__device__ __forceinline__ float leaky_relu(float x) {
    return x >= 0.0f ? x : ALPHA * x;
}

// Paired-K LDS layout: element (k, n) of the staged chunk lives at
//   s[(k/2)*RSTRIDE + 2*n + (k&1)]
// so the two K values a lane-half feeds into one WMMA are one aligned b64.
// Slot->column map per 8-column group kk: WMMA w, half hi, vgpr j
//   column = kk + 4*hi + 2*w + j
// (A and B use the same map; one global b128 covers both WMMAs of the group.)

// ---------------------------------------------------------------------------
// Kernel 1: Wh = h @ W   (8192x256 @ 256x64), v_wmma_f32_16x16x4_f32.
// ---------------------------------------------------------------------------
__global__ __launch_bounds__(256) void gat_wh_kernel(
        const float* __restrict__ h,
        const float* __restrict__ W,
        float* __restrict__ Wh) {
    __shared__ __align__(16) float sW[NROWS * RSTRIDE];   // 36 KB

    const int tid  = threadIdx.x;
    const int lane = tid & 31;
    const int wave = tid >> 5;
    const int hi   = (lane >> 4) & 1;
    const int nn   = lane & 15;
    const int r0   = (blockIdx.x * WAVES_PER_WG + wave) * ROWS_PER_WAVE;

    const float* hrow = h + (size_t)(r0 + nn) * F_IN;     // lane's A row (M = nn)

    v8f acc[4] = {};

    for (int kc = 0; kc < F_IN; kc += KC) {
        __syncthreads();
        {   // stage W[kc .. kc+KC) into paired-K layout (coalesced b32 reads, b64 LDS writes)
            const int n  = tid & 63;
            const int pp = tid >> 6;
            const float* src = W + (size_t)kc * F_OUT;
            for (int p = pp; p < NROWS; p += 4) {
                float2 v;
                v.x = src[(2 * p)     * F_OUT + n];
                v.y = src[(2 * p + 1) * F_OUT + n];
                *(float2*)&sW[p * RSTRIDE + 2 * n] = v;
            }
        }
        __syncthreads();

        for (int kk = 0; kk < KC; kk += 8) {
            // one b128 of h covers both WMMAs of this group
            const float4 av = *(const float4*)(hrow + kc + kk + 4 * hi);
            v2f aA; aA.x = av.x; aA.y = av.y;             // columns kk+4hi+{0,1}
            v2f aB; aB.x = av.z; aB.y = av.w;             // columns kk+4hi+{2,3}

            const int rb = ((kk >> 1) + 2 * hi) * RSTRIDE + 2 * nn;
#pragma unroll
            for (int t = 0; t < 4; ++t) {
                const v2f b = *(const v2f*)&sW[rb + 32 * t];
                acc[t] = __builtin_amdgcn_wmma_f32_16x16x4_f32(
                    false, aA, false, b, (short)0, acc[t], false, false);
            }
#pragma unroll
            for (int t = 0; t < 4; ++t) {
                const v2f b = *(const v2f*)&sW[rb + RSTRIDE + 32 * t];
                acc[t] = __builtin_amdgcn_wmma_f32_16x16x4_f32(
                    false, aB, false, b, (short)0, acc[t], false, false);
            }
        }
    }

    // C/D layout: VGPR j holds (M = j + 8*hi, N = 16*t + nn)
#pragma unroll
    for (int j = 0; j < 8; ++j) {
        const int row = r0 + j + 8 * hi;
#pragma unroll
        for (int t = 0; t < 4; ++t) {
            Wh[(size_t)row * F_OUT + 16 * t + nn] = acc[t][j];
        }
    }
}

// ---------------------------------------------------------------------------
// Kernel 1b: wh1[i] = Wh[i,:] . a[64:128],  wh2[i] = Wh[i,:] . a[0:64]
// ---------------------------------------------------------------------------
__global__ __launch_bounds__(256) void gat_wh12_kernel(
        const float* __restrict__ Wh,
        const float* __restrict__ a,
        float* __restrict__ wh1,
        float* __restrict__ wh2) {
    const int i = blockIdx.x * blockDim.x + threadIdx.x;
    if (i >= N_NODES) return;
    const float* r = Wh + (size_t)i * F_OUT;
    float s1 = 0.0f, s2 = 0.0f;
#pragma unroll 8
    for (int n = 0; n < F_OUT; ++n) {
        const float v = r[n];
        s1 += v * a[F_OUT + n];   // source (row) term uses a[f_out:]
        s2 += v * a[n];
    }
    wh1[i] = s1;
    wh2[i] = s2;
}

// ---------------------------------------------------------------------------
// Kernel 2: fused  e -> leakyrelu -> exp -> softmax-normalize -> @Wh -> ELU.
// Single streaming pass over adj (the 256 MB HBM roofline term). Non-edges
// contribute exp(0)=1, so the accumulation is dense and p feeds WMMA's A.
// ---------------------------------------------------------------------------
__global__ __launch_bounds__(256) void gat_attn_kernel(
        const float* __restrict__ adj,
        const float* __restrict__ Wh,
        const float* __restrict__ wh1,
        const float* __restrict__ wh2,
        float* __restrict__ out) {
    __shared__ __align__(16) float sWh[NROWS * RSTRIDE]; // 36 KB, shared by 8 waves
    __shared__ __align__(16) float sW2[KC];

    const int tid  = threadIdx.x;
    const int lane = tid & 31;
    const int wave = tid >> 5;
    const int hi   = (lane >> 4) & 1;
    const int nn   = lane & 15;
    const int r0   = (blockIdx.x * WAVES_PER_WG + wave) * ROWS_PER_WAVE;

    const float* adjrow = adj + (size_t)(r0 + nn) * N_NODES;  // lane's A row (M = nn)
    const float  w1     = wh1[r0 + nn];

    v8f   acc[4] = {};
    float ps     = 0.0f;                  // per-lane partial softmax denominator

    for (int k0 = 0; k0 < N_NODES; k0 += KC) {
        __syncthreads();
        {   // stage Wh[k0 .. k0+KC) into paired-K layout
            const int n  = tid & 63;
            const int pp = tid >> 6;
            const float* src = Wh + (size_t)k0 * F_OUT;
            for (int p = pp; p < NROWS; p += 4) {
                float2 v;
                v.x = src[(2 * p)     * F_OUT + n];
                v.y = src[(2 * p + 1) * F_OUT + n];
                *(float2*)&sWh[p * RSTRIDE + 2 * n] = v;
            }
        }
        if (tid < KC) sW2[tid] = wh2[k0 + tid];
        __syncthreads();

        // prefetch next adj chunk of this lane's row (global_prefetch_b8)
        if (k0 + KC < N_NODES)
            __builtin_prefetch(&adjrow[k0 + KC + 4 * hi], 0, 0);

        for (int kk = 0; kk < KC; kk += 8) {
            // adj + wh2 for this lane's 4 columns: kk + 4*hi + {0,1,2,3}
            const float4 av = *(const float4*)(adjrow + k0 + kk + 4 * hi);
            const float4 w2 = *(const float4*)&sW2[kk + 4 * hi];

            // p = adj!=0 ? exp(leakyrelu(wh1[u]+wh2[i])) : exp(0)=1
            const float p0 = (av.x != 0.0f) ? __expf(leaky_relu(w1 + w2.x)) : 1.0f;
            const float p1 = (av.y != 0.0f) ? __expf(leaky_relu(w1 + w2.y)) : 1.0f;
            const float p2 = (av.z != 0.0f) ? __expf(leaky_relu(w1 + w2.z)) : 1.0f;
            const float p3 = (av.w != 0.0f) ? __expf(leaky_relu(w1 + w2.w)) : 1.0f;
            ps += (p0 + p1) + (p2 + p3);

            v2f aA; aA.x = p0; aA.y = p1;
            v2f aB; aB.x = p2; aB.y = p3;

            const int rb = ((kk >> 1) + 2 * hi) * RSTRIDE + 2 * nn;
#pragma unroll
            for (int t = 0; t < 4; ++t) {
                const v2f b = *(const v2f*)&sWh[rb + 32 * t];
                acc[t] = __builtin_amdgcn_wmma_f32_16x16x4_f32(
                    false, aA, false, b, (short)0, acc[t], false, false);
            }
#pragma unroll
            for (int t = 0; t < 4; ++t) {
                const v2f b = *(const v2f*)&sWh[rb + RSTRIDE + 32 * t];
                acc[t] = __builtin_amdgcn_wmma_f32_16x16x4_f32(
                    false, aB, false, b, (short)0, acc[t], false, false);
            }
        }
    }

    // Lane l covers K%8 in {4hi..4hi+3} of row l&15; pair l^16 has the rest.
    const float rs = ps + __shfl_xor(ps, 16);

#pragma unroll
    for (int j = 0; j < 8; ++j) {
        const int   row = r0 + j + 8 * hi;
        const float den = __shfl(rs, j + 8 * hi);   // rowsum[j+8*hi] lives in that lane
        const float inv = 1.0f / den;
#pragma unroll
        for (int t = 0; t < 4; ++t) {
            float v = acc[t][j] * inv;
            v = (v > 0.0f) ? v : (__expf(v) - 1.0f);   // ELU
            out[(size_t)row * F_OUT + 16 * t + nn] = v;
        }
    }
}

// ---------------------------------------------------------------------------
extern "C" void kernel_launch(void* const* d_in, const int* in_sizes, int n_in,
                              void* d_out, int out_size, void* d_ws, size_t ws_size,
                              hipStream_t stream) {
    const float* h   = (const float*)d_in[0];   // [8192, 256]
    const float* adj = (const float*)d_in[1];   // [8192, 8192]
    const float* W   = (const float*)d_in[2];   // [256, 64]
    const float* a   = (const float*)d_in[3];   // [128, 1]
    float*       out = (float*)d_out;           // [8192, 64]

    float* Wh  = (float*)d_ws;                  // 8192*64 floats (2 MB)
    float* wh1 = Wh + (size_t)N_NODES * F_OUT;  // 8192 floats
    float* wh2 = wh1 + N_NODES;                 // 8192 floats

    gat_wh_kernel<<<dim3(N_NODES / (WAVES_PER_WG * ROWS_PER_WAVE)), dim3(256), 0, stream>>>(h, W, Wh);
    gat_wh12_kernel<<<dim3(N_NODES / 256), dim3(256), 0, stream>>>(Wh, a, wh1, wh2);
    gat_attn_kernel<<<dim3(N_NODES / (WAVES_PER_WG * ROWS_PER_WAVE)), dim3(256), 0, stream>>>(adj, Wh, wh1, wh2, out);
}